// MegalodonEMA_12455405158585
// MI455X (gfx1250) — compile-verified
//
#include <hip/hip_runtime.h>

typedef __attribute__((ext_vector_type(2))) float v2f;
typedef __attribute__((ext_vector_type(8))) float v8f;

#define LSEQ   2048
#define BSZ    8
#define DIM    1024
#define NDIM   64
#define CHUNK  16
#define NCHUNK (LSEQ / CHUNK)

static __device__ __forceinline__ v8f wmma4(v2f a, v2f b, v8f c) {
  // V_WMMA_F32_16X16X4_F32 : D = A(16x4 f32) * B(4x16 f32) + C(16x16 f32)
  return __builtin_amdgcn_wmma_f32_16x16x4_f32(
      false, a, false, b, (short)0, c, false, false);
}

static __device__ __forceinline__ float sigmoidf_(float v) {
  return 1.0f / (1.0f + __expf(-v));
}

__global__ __launch_bounds__(32)
void megalodon_ema_scan(const float* __restrict__ x,
                        const float* __restrict__ damping_factor,
                        const float* __restrict__ decay_factor,
                        const float* __restrict__ ema_exp,
                        const float* __restrict__ kproj,
                        const float* __restrict__ rweight,
                        float* __restrict__ out)
{
  __shared__ float sC[NDIM];           // c_n = damping*ema*kproj/sqrt(N)
  __shared__ float sLogW[NDIM];        // log(w_n)
  __shared__ float sDecay[NDIM];       // w_n^16 (per-chunk state decay)
  __shared__ float sK[CHUNK];          // kernel[0..15]
  __shared__ float sA[CHUNK * CHUNK];  // lower-tri Toeplitz (intra-chunk)
  __shared__ float sO[CHUNK * NDIM];   // O[l][n] = c_n * w_n^(l+1)
  __shared__ float sWin[NDIM * CHUNK]; // W_in[n][t] = w_n^(15-t)
  __shared__ float sX[CHUNK * 16];     // x tile, cols b>=8 stay zero forever
  __shared__ float sH[NDIM * 16];      // state h[n][b], b padded to 16

  const int lane = threadIdx.x;
  const int d    = blockIdx.x;
  const int col  = lane & 15;   // N index for B/C/D fragments, M index for A
  const int half = lane >> 4;   // lane half selects K-pair / M-upper-half

  // ---- phase 0: per-n scalars ----------------------------------------
  for (int n = lane; n < NDIM; n += 32) {
    float df = damping_factor[d * NDIM + n];
    float dc = decay_factor[d * NDIM + n];
    float damping = sigmoidf_(df);
    float decay   = sigmoidf_(dc);
    float w  = 1.0f - damping * decay;      // in (0,1)
    float lw = __logf(w);
    sLogW[n]  = lw;
    sC[n]     = damping * ema_exp[d * NDIM + n] * kproj[d * NDIM + n] * 0.125f;
    sDecay[n] = __expf(16.0f * lw);
  }
  __syncthreads();

  // ---- phase 1: per-d matrices in LDS --------------------------------
  if (lane < CHUNK) {
    float acc = 0.0f;
    for (int n = 0; n < NDIM; ++n)
      acc += sC[n] * __expf((float)lane * sLogW[n]);   // kernel[d, lane]
    sK[lane] = acc;
  }
  __syncthreads();
  for (int idx = lane; idx < CHUNK * CHUNK; idx += 32) {
    int i = idx >> 4, j = idx & 15;
    sA[idx] = (i >= j) ? sK[i - j] : 0.0f;
  }
  for (int idx = lane; idx < CHUNK * NDIM; idx += 32) {
    int l = idx >> 6, n = idx & 63;
    sO[idx] = sC[n] * __expf((float)(l + 1) * sLogW[n]);
  }
  for (int idx = lane; idx < NDIM * CHUNK; idx += 32) {
    int n = idx >> 4, t = idx & 15;
    sWin[idx] = __expf((float)(15 - t) * sLogW[n]);
  }
  for (int idx = lane; idx < NDIM * 16; idx += 32) sH[idx] = 0.0f;
  for (int idx = lane; idx < CHUNK * 16; idx += 32) sX[idx] = 0.0f;
  __syncthreads();

  // ---- phase 2: static A-operand fragments (live across chunk loop) --
  // A-matrix 16x4 f32 layout: element (m,k) -> lane = m + 16*(k>>1), vgpr = k&1
  v2f aA[4], aO[16], aW[16];
  #pragma unroll
  for (int f = 0; f < 4; ++f)
    #pragma unroll
    for (int j = 0; j < 2; ++j)
      aA[f][j] = sA[col * CHUNK + 4 * f + 2 * half + j];
  #pragma unroll
  for (int f = 0; f < 16; ++f)
    #pragma unroll
    for (int j = 0; j < 2; ++j)
      aO[f][j] = sO[col * NDIM + 4 * f + 2 * half + j];
  #pragma unroll
  for (int g = 0; g < 4; ++g)
    #pragma unroll
    for (int f = 0; f < 4; ++f)
      #pragma unroll
      for (int j = 0; j < 2; ++j)
        aW[g * 4 + f][j] = sWin[(16 * g + col) * CHUNK + 4 * f + 2 * half + j];

  const float rw = rweight[d];

  // Each lane owns batch column b = col (only b<8 lanes touch memory);
  // row within chunk for slot q is l = 2*q + half.
  float xr[8];
  if (col < BSZ) {
    #pragma unroll
    for (int q = 0; q < 8; ++q)
      xr[q] = x[(size_t)(2 * q + half) * (BSZ * DIM) + col * DIM + d];
  }

  // ---- phase 3: sequential chunk scan (double-buffered x) ------------
  for (int chunk = 0; chunk < NCHUNK; ++chunk) {
    const int cs = chunk * CHUNK;

    // commit prefetched x tile to LDS (padded cols already zero)
    if (col < BSZ) {
      #pragma unroll
      for (int q = 0; q < 8; ++q)
        sX[(2 * q + half) * 16 + col] = xr[q];
    }

    // issue next chunk's global loads now; s_wait lands after the WMMAs
    float xn[8];
    if (chunk + 1 < NCHUNK) {
      const int ncs = cs + CHUNK;
      if (col < BSZ) {
        #pragma unroll
        for (int q = 0; q < 8; ++q)
          xn[q] = x[(size_t)(ncs + 2 * q + half) * (BSZ * DIM) + col * DIM + d];
      }
    }
    __syncthreads();

    // B-matrix 4x16 f32 layout: element (k,n) -> lane = n + 16*(k>>1), vgpr = k&1
    v2f bX[4], bH[16];
    #pragma unroll
    for (int f = 0; f < 4; ++f)
      #pragma unroll
      for (int j = 0; j < 2; ++j)
        bX[f][j] = sX[(4 * f + 2 * half + j) * 16 + col];
    #pragma unroll
    for (int f = 0; f < 16; ++f)
      #pragma unroll
      for (int j = 0; j < 2; ++j)
        bH[f][j] = sH[(4 * f + 2 * half + j) * 16 + col];

    // y = A@x_chunk + O@h, split into two accumulators to halve the
    // WMMA->WMMA RAW chain (merged with 8 VALU adds at the end)
    v8f y0 = {0.f, 0.f, 0.f, 0.f, 0.f, 0.f, 0.f, 0.f};
    v8f y1 = {0.f, 0.f, 0.f, 0.f, 0.f, 0.f, 0.f, 0.f};
    #pragma unroll
    for (int f = 0; f < 4; ++f)  y0 = wmma4(aA[f], bX[f], y0);
    #pragma unroll
    for (int f = 0; f < 8; ++f) {
      y0 = wmma4(aO[2 * f],     bH[2 * f],     y0);
      y1 = wmma4(aO[2 * f + 1], bH[2 * f + 1], y1);
    }
    v8f y;
    #pragma unroll
    for (int r = 0; r < 8; ++r) y[r] = y0[r] + y1[r];

    // h' = diag(w^16) h + W_in @ x_chunk   (16 WMMAs, 4 independent chains)
    v8f hOut[4];
    #pragma unroll
    for (int g = 0; g < 4; ++g) {
      v8f hc;
      #pragma unroll
      for (int r = 0; r < 8; ++r) {
        int n = 16 * g + r + 8 * half;          // C/D layout row
        hc[r] = sH[n * 16 + col] * sDecay[n];
      }
      #pragma unroll
      for (int f = 0; f < 4; ++f) hc = wmma4(aW[g * 4 + f], bX[f], hc);
      hOut[g] = hc;
    }
    __syncthreads();
    #pragma unroll
    for (int g = 0; g < 4; ++g)
      #pragma unroll
      for (int r = 0; r < 8; ++r)
        sH[(16 * g + r + 8 * half) * 16 + col] = hOut[g][r];

    // out = silu(y + x*residual_weight); C/D layout: m = r + 8*half, b = col
    if (col < BSZ) {
      #pragma unroll
      for (int r = 0; r < 8; ++r) {
        int m = r + 8 * half;
        float v = y[r] + sX[m * 16 + col] * rw;
        out[(size_t)(cs + m) * (BSZ * DIM) + col * DIM + d] =
            v * (1.0f / (1.0f + __expf(-v)));
      }
    }
    __syncthreads();

    #pragma unroll
    for (int q = 0; q < 8; ++q) xr[q] = xn[q];
  }
}

extern "C" void kernel_launch(void* const* d_in, const int* in_sizes, int n_in,
                              void* d_out, int out_size, void* d_ws, size_t ws_size,
                              hipStream_t stream) {
  (void)in_sizes; (void)n_in; (void)out_size; (void)d_ws; (void)ws_size;
  const float* x  = (const float*)d_in[0];
  const float* df = (const float*)d_in[1];
  const float* dc = (const float*)d_in[2];
  const float* em = (const float*)d_in[3];
  const float* kp = (const float*)d_in[4];
  const float* rw = (const float*)d_in[5];
  float* out = (float*)d_out;

  megalodon_ema_scan<<<dim3(DIM), dim3(32), 0, stream>>>(x, df, dc, em, kp, rw, out);
}